// PrecomputedMetaNet_773094113739
// MI455X (gfx1250) — compile-verified
//
#include <hip/hip_runtime.h>
#include <stdint.h>

// CDNA5 (gfx1250) fp32 WMMA implementation of PrecomputedMetaNet.
//
// Roofline: ~80 GFLOP fp32, ~64 MB unique HBM traffic (2.8 us @ 23.3 TB/s);
// intermediates (16 MB ping-pong, 4 MB h) live in the 192 MB L2. Compute
// bound on the matrix pipe -> V_WMMA_F32_16X16X4_F32 (full fp32 precision).
// Global->LDS staging uses GLOBAL_LOAD_ASYNC_TO_LDS_B128 with double-buffered
// LDS (ASYNCcnt-tracked), so tile movement overlaps the WMMA stream.

typedef float v2f __attribute__((ext_vector_type(2)));
typedef float v8f __attribute__((ext_vector_type(8)));

#define BM 128
#define BN 128
#define BK 16

// LDS layouts (words):
//  A      : m-major [128][LAW], fragment = contiguous (k,k+1) pair -> ds_load_b64
//  B (BT) : n-major [128][LBN], fragment = contiguous pair -> ds_load_b64
//  B (ROW): k-major [BK][LBT], fragment = strided pair (global layout is k-major)
#define LAW 20
#define LBN 20
#define LBT 136
#define A_WORDS (BM * LAW)         // 2560
#define B_WORDS 2560               // max(BK*LBT=2176, BM*LBN=2560)

#define DIM_D 1024
#define DIM_H 256
#define DIM_T 8
#define DIM_B 4096

enum { MODE_RELU = 0, MODE_TASK = 1, MODE_PLAIN = 2 };
// MODE_RELU : D = relu(A @ Bsrc^T + bias)        (Bsrc = W1 [N,K] row-major)
// MODE_TASK : D = A + coeff[:,j] * (A @ Bsrc)    (Bsrc = M_j [K,N] row-major, N==K)
// MODE_PLAIN: D = A @ Bsrc^T                     (Bsrc = Wp [N,K] row-major)

__device__ __forceinline__ void async_ld_b128(uint32_t lds_byte,
                                              uint32_t glb_off,
                                              const void* base) {
  // GVS mode: reads global memory at the 64-bit base in an SGPR plus offset glb_off, writes LDS at lds_byte.
  asm volatile("global_load_async_to_lds_b128 %0, %1, %2"
               :
               : "v"(lds_byte), "v"(glb_off), "s"(base)
               : "memory");
}

template <int MODE>
__global__ __launch_bounds__(256) void wmma_gemm_f32(
    const float* __restrict__ A, const float* __restrict__ Bsrc,
    float* __restrict__ Dout, const float* __restrict__ bias,
    const float* __restrict__ coeff, int jidx, int M, int N, int K) {
  __shared__ __align__(16) float lA[2 * A_WORDS];
  __shared__ __align__(16) float lB[2 * B_WORDS];

  const int tid = threadIdx.x;
  const int lane = tid & 31;
  const int wave = tid >> 5;     // 8 waves (wave32)
  const int wm = wave >> 1;      // 0..3 -> 32 rows each
  const int wn = wave & 1;       // 0..1 -> 64 cols each
  const int half = lane >> 4;    // K-half selector per ISA A/B layout
  const int l16 = lane & 15;

  const int bm0 = blockIdx.y * BM;
  const int bn0 = blockIdx.x * BN;

  const uint32_t aBase = (uint32_t)(uintptr_t)&lA[0];
  const uint32_t bBase = (uint32_t)(uintptr_t)&lB[0];

  // Issue one K-slab's async tile loads (4 x b128 per thread-pair pattern,
  // 4 async instructions per wave -> ASYNCcnt +4 per slab).
  auto stage = [&](int k0s, int bufsel) {
#pragma unroll
    for (int q = 0; q < 2; ++q) {  // A tile: 128x16, m-major in LDS
      const int idx = tid * 2 + q;
      const int m = idx >> 2;
      const int kq = idx & 3;
      const uint32_t lds =
          aBase + (uint32_t)(bufsel * (A_WORDS * 4) + m * (LAW * 4) + kq * 16);
      const uint32_t go =
          (uint32_t)(((bm0 + m) * K + k0s + kq * 4) * sizeof(float));
      async_ld_b128(lds, go, A);
    }
    if (MODE == MODE_TASK) {  // B tile 16x128, k-major source & LDS
#pragma unroll
      for (int q = 0; q < 2; ++q) {
        const int idx = tid * 2 + q;
        const int r = idx >> 5;   // k row 0..15
        const int c4 = idx & 31;  // col quad
        const uint32_t lds =
            bBase +
            (uint32_t)(bufsel * (B_WORDS * 4) + r * (LBT * 4) + c4 * 16);
        const uint32_t go =
            (uint32_t)(((k0s + r) * N + bn0 + c4 * 4) * sizeof(float));
        async_ld_b128(lds, go, Bsrc);
      }
    } else {  // transposed source: rows of Bsrc are n-major; keep n-major LDS
#pragma unroll
      for (int q = 0; q < 2; ++q) {
        const int idx = tid * 2 + q;
        const int n = idx >> 2;
        const int kq = idx & 3;
        const uint32_t lds =
            bBase +
            (uint32_t)(bufsel * (B_WORDS * 4) + n * (LBN * 4) + kq * 16);
        const uint32_t go =
            (uint32_t)(((bn0 + n) * K + k0s + kq * 4) * sizeof(float));
        async_ld_b128(lds, go, Bsrc);
      }
    }
  };

  v8f acc[2][4];
#pragma unroll
  for (int tm = 0; tm < 2; ++tm)
#pragma unroll
    for (int tn = 0; tn < 4; ++tn) acc[tm][tn] = v8f{};

  stage(0, 0);

  for (int k0 = 0; k0 < K; k0 += BK) {
    const int cur = (k0 >> 4) & 1;
    if (k0 + BK < K) {
      stage(k0 + BK, cur ^ 1);  // prefetch next slab into other buffer
      asm volatile("s_wait_asynccnt 0x4" ::: "memory");  // cur slab landed
    } else {
      asm volatile("s_wait_asynccnt 0x0" ::: "memory");
    }
    __syncthreads();

    const float* lAc = lA + cur * A_WORDS;
    const float* lBc = lB + cur * B_WORDS;

#pragma unroll
    for (int kk = 0; kk < BK; kk += 4) {
      const int kb = kk + half * 2;  // lanes 0-15: K=kk,kk+1 ; 16-31: +2
      v2f afrag[2];
#pragma unroll
      for (int tm = 0; tm < 2; ++tm) {
        const int m = wm * 32 + tm * 16 + l16;
        afrag[tm] = *(const v2f*)(lAc + m * LAW + kb);  // ds_load_b64
      }
      v2f bfrag[4];
#pragma unroll
      for (int tn = 0; tn < 4; ++tn) {
        const int n = wn * 64 + tn * 16 + l16;
        if (MODE == MODE_TASK) {
          bfrag[tn].x = lBc[kb * LBT + n];
          bfrag[tn].y = lBc[(kb + 1) * LBT + n];
        } else {
          bfrag[tn] = *(const v2f*)(lBc + n * LBN + kb);  // ds_load_b64
        }
      }
#pragma unroll
      for (int tm = 0; tm < 2; ++tm)
#pragma unroll
        for (int tn = 0; tn < 4; ++tn)
          acc[tm][tn] = __builtin_amdgcn_wmma_f32_16x16x4_f32(
              false, afrag[tm], false, bfrag[tn], (short)0, acc[tm][tn],
              false, false);
    }
    __syncthreads();  // all reads done before this buffer is re-staged
  }

  // ---- epilogue: C/D layout row M = r + 8*half, col N = l16 ----
#pragma unroll
  for (int tm = 0; tm < 2; ++tm) {
    const int mbase = bm0 + wm * 32 + tm * 16 + half * 8;
    float cvals[8];
    if (MODE == MODE_TASK) {
#pragma unroll
      for (int r = 0; r < 8; ++r)
        cvals[r] = coeff[(size_t)(mbase + r) * DIM_T + jidx];
    }
#pragma unroll
    for (int tn = 0; tn < 4; ++tn) {
      const int n = bn0 + wn * 64 + tn * 16 + l16;
#pragma unroll
      for (int r = 0; r < 8; ++r) {
        const int m = mbase + r;
        float v = acc[tm][tn][r];
        if (MODE == MODE_RELU) {
          v += bias[n];
          v = v > 0.f ? v : 0.f;
        } else if (MODE == MODE_TASK) {
          // N == K for the task steps, so A[m][n] is the residual input.
          v = A[(size_t)m * K + n] + cvals[r] * v;
        }
        Dout[(size_t)m * N + n] = v;
      }
    }
  }
}

// coeff[b][t] = b2[t] + sum_h h[b][h] * W2[t][h]   (4096x8x256, trivial)
__global__ __launch_bounds__(256) void coeff_kernel(
    const float* __restrict__ h, const float* __restrict__ W2,
    const float* __restrict__ b2, float* __restrict__ coeff) {
  const int idx = blockIdx.x * 256 + threadIdx.x;  // b*T + t
  const int b = idx >> 3;
  const int t = idx & 7;
  const float4* hr = (const float4*)(h + (size_t)b * DIM_H);
  const float4* wr = (const float4*)(W2 + (size_t)t * DIM_H);
  float s = b2[t];
#pragma unroll 4
  for (int k = 0; k < DIM_H / 4; ++k) {
    const float4 a = hr[k], w = wr[k];
    s += a.x * w.x + a.y * w.y + a.z * w.z + a.w * w.w;
  }
  coeff[idx] = s;
}

extern "C" void kernel_launch(void* const* d_in, const int* in_sizes, int n_in,
                              void* d_out, int out_size, void* d_ws,
                              size_t ws_size, hipStream_t stream) {
  const float* features = (const float*)d_in[0];
  const float* W1 = (const float*)d_in[1];
  const float* b1 = (const float*)d_in[2];
  const float* W2 = (const float*)d_in[3];
  const float* b2 = (const float*)d_in[4];
  const float* task = (const float*)d_in[5];
  const float* Wp = (const float*)d_in[6];
  float* out = (float*)d_out;

  // ws layout: h (4 MB) | coeff (128 KB) | ping buffer (16 MB) ~= 20.1 MB
  float* h = (float*)d_ws;
  float* coeff = h + (size_t)DIM_B * DIM_H;
  float* buf = coeff + (size_t)DIM_B * DIM_T;

  const dim3 blk(256);

  // 1) h = relu(F @ W1^T + b1)
  wmma_gemm_f32<MODE_RELU><<<dim3(DIM_H / BN, DIM_B / BM), blk, 0, stream>>>(
      features, W1, h, b1, nullptr, 0, DIM_B, DIM_H, DIM_D);

  // 2) coeff = h @ W2^T + b2
  coeff_kernel<<<(DIM_B * DIM_T) / 256, blk, 0, stream>>>(h, W2, b2, coeff);

  // 3) eight chained task steps; even steps -> d_out, odd steps -> ws buf
  const float* src = features;
  for (int j = 0; j < DIM_T; ++j) {
    float* dst = (j & 1) ? buf : out;
    wmma_gemm_f32<MODE_TASK><<<dim3(DIM_D / BN, DIM_B / BM), blk, 0, stream>>>(
        src, task + (size_t)j * DIM_D * DIM_D, dst, nullptr, coeff, j, DIM_B,
        DIM_D, DIM_D);
    src = dst;  // step 7 (odd) ends in buf
  }

  // 4) out = t @ Wp^T
  wmma_gemm_f32<MODE_PLAIN><<<dim3(DIM_D / BN, DIM_B / BM), blk, 0, stream>>>(
      buf, Wp, out, nullptr, nullptr, 0, DIM_B, DIM_D, DIM_D);
}